// Net_89464168776168
// MI455X (gfx1250) — compile-verified
//
#include <hip/hip_runtime.h>
#include <hip/hip_bf16.h>
#include <math.h>

typedef float v2f __attribute__((ext_vector_type(2)));
typedef float v8f __attribute__((ext_vector_type(8)));

#define BATCH 8192

// ---------------------------------------------------------------------------
// Bilinear grid sample with zeros padding (matches reference semantics).
// ---------------------------------------------------------------------------
__device__ __forceinline__ float gs_fetch(const float* img, int y, int x, int H, int W) {
    if (x < 0 || x >= W || y < 0 || y >= H) return 0.0f;
    return img[y * W + x];
}

__device__ __forceinline__ float grid_sample_zeros(const float* img, float gx, float gy,
                                                   int H, int W) {
    float ix = ((gx + 1.0f) * (float)W - 1.0f) * 0.5f;
    float iy = ((gy + 1.0f) * (float)H - 1.0f) * 0.5f;
    float x0f = floorf(ix), y0f = floorf(iy);
    int   x0 = (int)x0f,    y0 = (int)y0f;
    float wx1 = ix - x0f, wx0 = 1.0f - wx1;
    float wy1 = iy - y0f, wy0 = 1.0f - wy1;
    float v00 = gs_fetch(img, y0,     x0,     H, W);
    float v01 = gs_fetch(img, y0,     x0 + 1, H, W);
    float v10 = gs_fetch(img, y0 + 1, x0,     H, W);
    float v11 = gs_fetch(img, y0 + 1, x0 + 1, H, W);
    return v00 * wy0 * wx0 + v01 * wy0 * wx1 + v10 * wy1 * wx0 + v11 * wy1 * wx1;
}

// ---------------------------------------------------------------------------
// Layer 1: 1->6 ch, 5x5 conv pad 1 (26 -> 24), deformable sample, relu, 2x2 pool
// One workgroup per image. Everything staged in LDS.
// ---------------------------------------------------------------------------
__global__ void __launch_bounds__(256) layer1_kernel(
    const float* __restrict__ x,
    const float* __restrict__ w1s, const float* __restrict__ b1s,
    const float* __restrict__ w1x, const float* __restrict__ b1x,
    const float* __restrict__ w1y, const float* __restrict__ b1y,
    float* __restrict__ out /* B x 6 x 12 x 12 */) {
    __shared__ float simg[26 * 26];
    __shared__ float ss[6 * 24 * 24];
    __shared__ float sw[3 * 6 * 25];
    __shared__ float sb[3 * 6];

    const int b   = blockIdx.x;
    const int tid = threadIdx.x;
    const float* xim = x + (size_t)b * (26 * 26);

    for (int i = tid; i < 26 * 26; i += blockDim.x) simg[i] = xim[i];
    for (int i = tid; i < 150; i += blockDim.x) {
        sw[i]       = w1s[i];
        sw[150 + i] = w1x[i];
        sw[300 + i] = w1y[i];
    }
    if (tid < 6) { sb[tid] = b1s[tid]; sb[6 + tid] = b1x[tid]; sb[12 + tid] = b1y[tid]; }
    __syncthreads();

    // s-conv into LDS
    for (int idx = tid; idx < 6 * 24 * 24; idx += blockDim.x) {
        int c  = idx / (24 * 24);
        int p  = idx % (24 * 24);
        int oy = p / 24, ox = p % 24;
        float acc = sb[c];
        const float* wc = sw + c * 25;
#pragma unroll
        for (int ky = 0; ky < 5; ++ky) {
            int iy = oy - 1 + ky;
            if ((unsigned)iy >= 26u) continue;
#pragma unroll
            for (int kx = 0; kx < 5; ++kx) {
                int ix = ox - 1 + kx;
                if ((unsigned)ix >= 26u) continue;
                acc = fmaf(wc[ky * 5 + kx], simg[iy * 26 + ix], acc);
            }
        }
        ss[idx] = acc;
    }
    __syncthreads();

    // grid sample + relu + maxpool
    for (int idx = tid; idx < 6 * 12 * 12; idx += blockDim.x) {
        int c  = idx / 144;
        int p  = idx % 144;
        int py = p / 12, px = p % 12;
        const float* wxc = sw + 150 + c * 25;
        const float* wyc = sw + 300 + c * 25;
        float m = 0.0f;  // relu outputs are >= 0
#pragma unroll
        for (int dy = 0; dy < 2; ++dy) {
#pragma unroll
            for (int dx = 0; dx < 2; ++dx) {
                int oy = py * 2 + dy, ox = px * 2 + dx;
                float gx = sb[6 + c], gy = sb[12 + c];
#pragma unroll
                for (int ky = 0; ky < 5; ++ky) {
                    int iy = oy - 1 + ky;
                    if ((unsigned)iy >= 26u) continue;
#pragma unroll
                    for (int kx = 0; kx < 5; ++kx) {
                        int ix = ox - 1 + kx;
                        if ((unsigned)ix >= 26u) continue;
                        float v = simg[iy * 26 + ix];
                        gx = fmaf(wxc[ky * 5 + kx], v, gx);
                        gy = fmaf(wyc[ky * 5 + kx], v, gy);
                    }
                }
                float v = grid_sample_zeros(ss + c * 576, gx, gy, 24, 24);
                v = fmaxf(v, 0.0f);
                m = fmaxf(m, v);
            }
        }
        out[(size_t)b * 864 + idx] = m;
    }
}

// ---------------------------------------------------------------------------
// Layer 2: 6->16 ch, 5x5 conv pad 1 (12 -> 10), deformable sample, relu, pool
// ---------------------------------------------------------------------------
__global__ void __launch_bounds__(256) layer2_kernel(
    const float* __restrict__ in /* B x 6 x 12 x 12 */,
    const float* __restrict__ w2s, const float* __restrict__ b2s,
    const float* __restrict__ w2x, const float* __restrict__ b2x,
    const float* __restrict__ w2y, const float* __restrict__ b2y,
    float* __restrict__ out /* B x 16 x 5 x 5 */) {
    __shared__ float simg[6 * 12 * 12];   // 864
    __shared__ float ss[16 * 10 * 10];    // 1600
    __shared__ float sw[3 * 16 * 6 * 25]; // 7200
    __shared__ float sb[3 * 16];

    const int b   = blockIdx.x;
    const int tid = threadIdx.x;
    const float* xim = in + (size_t)b * 864;

    for (int i = tid; i < 864; i += blockDim.x) simg[i] = xim[i];
    for (int i = tid; i < 2400; i += blockDim.x) {
        sw[i]        = w2s[i];
        sw[2400 + i] = w2x[i];
        sw[4800 + i] = w2y[i];
    }
    if (tid < 16) { sb[tid] = b2s[tid]; sb[16 + tid] = b2x[tid]; sb[32 + tid] = b2y[tid]; }
    __syncthreads();

    // s-conv into LDS
    for (int idx = tid; idx < 16 * 10 * 10; idx += blockDim.x) {
        int c  = idx / 100;
        int p  = idx % 100;
        int oy = p / 10, ox = p % 10;
        float acc = sb[c];
        for (int ci = 0; ci < 6; ++ci) {
            const float* wc = sw + (c * 6 + ci) * 25;
            const float* ic = simg + ci * 144;
#pragma unroll
            for (int ky = 0; ky < 5; ++ky) {
                int iy = oy - 1 + ky;
                if ((unsigned)iy >= 12u) continue;
#pragma unroll
                for (int kx = 0; kx < 5; ++kx) {
                    int ix = ox - 1 + kx;
                    if ((unsigned)ix >= 12u) continue;
                    acc = fmaf(wc[ky * 5 + kx], ic[iy * 12 + ix], acc);
                }
            }
        }
        ss[idx] = acc;
    }
    __syncthreads();

    // grid sample + relu + maxpool -> 16 x 5 x 5
    for (int idx = tid; idx < 16 * 5 * 5; idx += blockDim.x) {
        int c  = idx / 25;
        int p  = idx % 25;
        int py = p / 5, px = p % 5;
        float m = 0.0f;
#pragma unroll
        for (int dy = 0; dy < 2; ++dy) {
#pragma unroll
            for (int dx = 0; dx < 2; ++dx) {
                int oy = py * 2 + dy, ox = px * 2 + dx;
                float gx = sb[16 + c], gy = sb[32 + c];
                for (int ci = 0; ci < 6; ++ci) {
                    const float* wxc = sw + 2400 + (c * 6 + ci) * 25;
                    const float* wyc = sw + 4800 + (c * 6 + ci) * 25;
                    const float* ic  = simg + ci * 144;
#pragma unroll
                    for (int ky = 0; ky < 5; ++ky) {
                        int iy = oy - 1 + ky;
                        if ((unsigned)iy >= 12u) continue;
#pragma unroll
                        for (int kx = 0; kx < 5; ++kx) {
                            int ix = ox - 1 + kx;
                            if ((unsigned)ix >= 12u) continue;
                            float v = ic[iy * 12 + ix];
                            gx = fmaf(wxc[ky * 5 + kx], v, gx);
                            gy = fmaf(wyc[ky * 5 + kx], v, gy);
                        }
                    }
                }
                float v = grid_sample_zeros(ss + c * 100, gx, gy, 10, 10);
                v = fmaxf(v, 0.0f);
                m = fmaxf(m, v);
            }
        }
        out[(size_t)b * 400 + idx] = m;
    }
}

// ---------------------------------------------------------------------------
// FC GEMM on the matrix pipe: out[M,N] = A[M,K] * W[N,K]^T + bias, opt. relu.
// fp32 in / fp32 out via V_WMMA_F32_16X16X4_F32 (keeps full precision; this
// workload is latency-bound, not matrix-peak-bound, so fp32 WMMA is the right
// precision choice on MI455X).
//
// Fragment layout (wave32, ISA 7.12.2):
//   A 16x4 f32:  lane l<16 holds A[m0+l][k+0..1], lane l>=16 holds K=2..3.
//   B 4x16 f32:  mirrored — lane l<16 holds W[n0+l][k+0..1] (= B[k][n0+l]).
//   C/D 16x16:   vgpr i, lanes 0-15 -> row i, lanes 16-31 -> row i+8.
// One wave per 16x16 C tile; 8 waves/block cover the N tiles of one M stripe.
// ---------------------------------------------------------------------------
__global__ void __launch_bounds__(256) wmma_fc_kernel(
    const float* __restrict__ A, const float* __restrict__ W,
    const float* __restrict__ bias, float* __restrict__ out,
    int M, int N, int K, int relu) {
    const int m0     = blockIdx.x * 16;
    const int lane   = threadIdx.x & 31;
    const int wave   = threadIdx.x >> 5;
    const int nwaves = blockDim.x >> 5;
    const int half   = lane >> 4;   // 0: K pair {0,1}, 1: K pair {2,3}
    const int l16    = lane & 15;

    const float* Arow = A + (size_t)(m0 + l16) * K + half * 2;

    for (int nt = wave; nt * 16 < N; nt += nwaves) {
        const int  n0     = nt * 16;
        const int  ncol   = n0 + l16;
        const bool nvalid = (ncol < N);
        // Clamp to stay in bounds; garbage columns are never stored.
        const int  wrow   = nvalid ? ncol : (N - 1);
        const float* Wrow = W + (size_t)wrow * K + half * 2;

        v8f c = {};
        for (int k = 0; k < K; k += 4) {
            v2f a = *(const v2f*)(Arow + k);
            v2f bfrag = *(const v2f*)(Wrow + k);
            c = __builtin_amdgcn_wmma_f32_16x16x4_f32(
                    false, a, false, bfrag, (short)0, c, false, false);
        }

        if (nvalid) {
            const float bv = bias[ncol];
#pragma unroll
            for (int i = 0; i < 8; ++i) {
                int   row = m0 + i + half * 8;
                float v   = c[i] + bv;
                if (relu) v = fmaxf(v, 0.0f);
                out[(size_t)row * N + ncol] = v;
            }
        }
    }
}

// ---------------------------------------------------------------------------
// Softmax over 10 logits per row.
// ---------------------------------------------------------------------------
__global__ void __launch_bounds__(256) softmax10_kernel(
    const float* __restrict__ lg, float* __restrict__ out, int M) {
    int i = blockIdx.x * blockDim.x + threadIdx.x;
    if (i >= M) return;
    const float* r = lg + (size_t)i * 10;
    float mx = r[0];
#pragma unroll
    for (int j = 1; j < 10; ++j) mx = fmaxf(mx, r[j]);
    float e[10], s = 0.0f;
#pragma unroll
    for (int j = 0; j < 10; ++j) { e[j] = expf(r[j] - mx); s += e[j]; }
    float inv = 1.0f / s;
#pragma unroll
    for (int j = 0; j < 10; ++j) out[(size_t)i * 10 + j] = e[j] * inv;
}

// ---------------------------------------------------------------------------
extern "C" void kernel_launch(void* const* d_in, const int* in_sizes, int n_in,
                              void* d_out, int out_size, void* d_ws, size_t ws_size,
                              hipStream_t stream) {
    const float* x   = (const float*)d_in[0];
    const float* w1s = (const float*)d_in[1];
    const float* b1s = (const float*)d_in[2];
    const float* w1x = (const float*)d_in[3];
    const float* b1x = (const float*)d_in[4];
    const float* w1y = (const float*)d_in[5];
    const float* b1y = (const float*)d_in[6];
    const float* w2s = (const float*)d_in[7];
    const float* b2s = (const float*)d_in[8];
    const float* w2x = (const float*)d_in[9];
    const float* b2x = (const float*)d_in[10];
    const float* w2y = (const float*)d_in[11];
    const float* b2y = (const float*)d_in[12];
    const float* wf1 = (const float*)d_in[13];
    const float* bf1 = (const float*)d_in[14];
    const float* wf2 = (const float*)d_in[15];
    const float* bf2 = (const float*)d_in[16];
    const float* wf3 = (const float*)d_in[17];
    const float* bf3 = (const float*)d_in[18];
    float* out = (float*)d_out;

    float* ws = (float*)d_ws;
    float* p1 = ws;                         // B x 864
    float* p2 = p1 + (size_t)BATCH * 864;   // B x 400
    float* h1 = p2 + (size_t)BATCH * 400;   // B x 120
    float* h2 = h1 + (size_t)BATCH * 120;   // B x 84
    float* lg = h2 + (size_t)BATCH * 84;    // B x 10

    layer1_kernel<<<BATCH, 256, 0, stream>>>(x, w1s, b1s, w1x, b1x, w1y, b1y, p1);
    layer2_kernel<<<BATCH, 256, 0, stream>>>(p1, w2s, b2s, w2x, b2x, w2y, b2y, p2);

    // FC1: 400 -> 120, relu     (8 N-tiles -> 8 waves/block)
    wmma_fc_kernel<<<BATCH / 16, 256, 0, stream>>>(p2, wf1, bf1, h1, BATCH, 120, 400, 1);
    // FC2: 120 -> 84, relu      (6 N-tiles)
    wmma_fc_kernel<<<BATCH / 16, 256, 0, stream>>>(h1, wf2, bf2, h2, BATCH, 84, 120, 1);
    // FC3: 84 -> 10             (1 N-tile -> single wave per block)
    wmma_fc_kernel<<<BATCH / 16, 32, 0, stream>>>(h2, wf3, bf3, lg, BATCH, 10, 84, 0);

    softmax10_kernel<<<(BATCH + 255) / 256, 256, 0, stream>>>(lg, out, BATCH);
}